// Resample2d_34187939676300
// MI455X (gfx1250) — compile-verified
//
#include <hip/hip_runtime.h>

// FlowNet2 resample2d (bilinear warp), fp32.
// Shapes fixed by reference setup_inputs(): B=4, C=64, H=384, W=512.
//
// Memory-bound: ~0.41 GB moved, ~0.5 GFLOP -> AI ~1.2 F/B; floor ~18us at
// 23.3 TB/s. Strategy: 1 thread = 4 consecutive x pixels (b128 NT stores),
// 2 x b64 gathers per pixel per channel (L/R taps adjacent in memory), the
// edge-clamp tap selection folded into per-pixel weights, and all indexing
// kept as (uniform pointer + divergent i32) so loads/stores lower to the
// SADDR form: zero per-channel address VALU, one SALU base bump per channel.

constexpr int B = 4, C = 64, H = 384, W = 512;
constexpr int HW = H * W;
constexpr int PPT = 4;                        // x-pixels per thread
constexpr int BLOCK = 256;
constexpr int ROWS_PER_BLOCK = BLOCK * PPT / W;          // 2
constexpr int BLOCKS_PER_BATCH = H / ROWS_PER_BLOCK;     // 192
constexpr int NBLOCKS = B * BLOCKS_PER_BATCH;            // 768

typedef float v2f __attribute__((ext_vector_type(2)));
typedef float v4f __attribute__((ext_vector_type(4)));
typedef v2f v2f_u __attribute__((aligned(4)));   // 4B-aligned 8B vector load

__device__ __forceinline__ v2f load2(const float* p) {
    return *reinterpret_cast<const v2f_u*>(p);
}
__device__ __forceinline__ int imin(int a, int b) { return a < b ? a : b; }
__device__ __forceinline__ int imax(int a, int b) { return a > b ? a : b; }

__global__ __launch_bounds__(BLOCK) void resample2d_warp(
    const float* __restrict__ in1, const float* __restrict__ flow,
    float* __restrict__ out)
{
    // Block-uniform (SALU) decomposition: each block = 2 rows of one batch.
    int b       = (int)blockIdx.x / BLOCKS_PER_BATCH;
    int rowbase = ((int)blockIdx.x % BLOCKS_PER_BATCH) * ROWS_PER_BLOCK;
    // Per-thread (VALU): row within block + column. All offsets fit in i32.
    int h   = rowbase + ((int)threadIdx.x >> 7);     // 128 threads per row
    int w   = ((int)threadIdx.x & 127) << 2;         // PPT = 4
    int pix = h * W + w;                             // divergent, invariant

    // Uniform batch bases (SALU): b is uniform per block.
    const float* __restrict__ flowb = flow + b * (2 * HW);
    const float* __restrict__ srcb  = in1  + b * (C * HW);
    float*       __restrict__ dstb  = out  + b * (C * HW);

    // Flow: [B, 2, H, W]; dx plane then dy plane. 16B-aligned (w % 4 == 0).
    v4f dx4 = __builtin_nontemporal_load(reinterpret_cast<const v4f*>(flowb + pix));
    v4f dy4 = __builtin_nontemporal_load(reinterpret_cast<const v4f*>(flowb + HW + pix));

    int   offT[PPT], offB[PPT];      // top/bottom row element offsets (at xb)
    float wTx[PPT], wTy[PPT], wBx[PPT], wBy[PPT];   // clamp-folded weights

    float dxs[PPT] = {dx4.x, dx4.y, dx4.z, dx4.w};
    float dys[PPT] = {dy4.x, dy4.y, dy4.z, dy4.w};

#pragma unroll
    for (int i = 0; i < PPT; ++i) {
        float xf = (float)(w + i) + dxs[i];
        float yf = (float)h + dys[i];
        float x0 = floorf(xf), y0 = floorf(yf);
        float a  = xf - x0,  be = yf - y0;       // alpha, beta (unclamped)
        int x0i = (int)x0, y0i = (int)y0;
        int xL = imin(imax(x0i,     0), W - 1);
        int xR = imin(imax(x0i + 1, 0), W - 1);
        int yT = imin(imax(y0i,     0), H - 1);
        int yB = imin(imax(y0i + 1, 0), H - 1);
        int xb = imin(imax(x0i,     0), W - 2);  // base column of the b64 tap
        // Tap selection is channel-invariant -> fold it into the weights:
        //   value(xL) = sL ? v.y : v.x ; value(xR) = sR ? v.y : v.x
        bool sL = (xL != xb);
        bool sR = (xR != xb);
        offT[i] = yT * W + xb;
        offB[i] = yB * W + xb;
        float ia = 1.0f - a, ib = 1.0f - be;
        float wTL = ia * ib, wTR = a * ib, wBL = ia * be, wBR = a * be;
        wTx[i] = (sL ? 0.0f : wTL) + (sR ? 0.0f : wTR);
        wTy[i] = (sL ? wTL : 0.0f) + (sR ? wTR : 0.0f);
        wBx[i] = (sL ? 0.0f : wBL) + (sR ? 0.0f : wBR);
        wBy[i] = (sL ? wBL : 0.0f) + (sR ? wBR : 0.0f);
    }

    // Warm the first gather lines (gfx1250 global_prefetch_b8).
    __builtin_prefetch(srcb + offT[0],       0, 0);
    __builtin_prefetch(srcb + offB[0],       0, 0);
    __builtin_prefetch(srcb + offT[PPT - 1], 0, 0);
    __builtin_prefetch(srcb + offB[PPT - 1], 0, 0);

#pragma unroll 2
    for (int c = 0; c < C; ++c) {
        const float* __restrict__ sc = srcb + c * HW;   // uniform base
        float res[PPT];
#pragma unroll
        for (int i = 0; i < PPT; ++i) {
            v2f vT = load2(sc + offT[i]);   // top row:  (xb, xb+1)
            v2f vB = load2(sc + offB[i]);   // bot row:  (xb, xb+1)
            res[i] = wTx[i] * vT.x + wTy[i] * vT.y
                   + wBx[i] * vB.x + wBy[i] * vB.y;
        }
        v4f o = {res[0], res[1], res[2], res[3]};
        // Output is write-once: NT store keeps the 192MB L2 for gather reuse.
        __builtin_nontemporal_store(
            o, reinterpret_cast<v4f*>(dstb + c * HW + pix));
    }
}

extern "C" void kernel_launch(void* const* d_in, const int* in_sizes, int n_in,
                              void* d_out, int out_size, void* d_ws, size_t ws_size,
                              hipStream_t stream) {
    const float* in1  = (const float*)d_in[0];   // [B, C, H, W] fp32
    const float* flow = (const float*)d_in[1];   // [B, 2, H, W] fp32
    float* out = (float*)d_out;                  // [B, C, H, W] fp32
    resample2d_warp<<<dim3(NBLOCKS), dim3(BLOCK), 0, stream>>>(in1, flow, out);
}